// HungarianMatcher_84490596647596
// MI455X (gfx1250) — compile-verified
//
#include <hip/hip_runtime.h>

#define BS   16
#define Q    1024
#define NCLS 91
#define PK   96      // NCLS padded to multiple of 4 (and 32) for the WMMA K loop
#define T    128

typedef float v2f __attribute__((ext_vector_type(2)));
typedef float v8f __attribute__((ext_vector_type(8)));

// ---------------------------------------------------------------------------
// Kernel 1: row softmax of logits [BS*Q, 91] -> padded probs [BS*Q, 96]
// One wave32 per row; 8 waves (256 threads) per block.
// ---------------------------------------------------------------------------
__global__ void hm_softmax_rows(const float* __restrict__ logits,
                                float* __restrict__ probs) {
    const int lane = threadIdx.x & 31;
    const int wave = threadIdx.x >> 5;
    const int row  = blockIdx.x * 8 + wave;          // 0 .. BS*Q-1

    const float* src = logits + (size_t)row * NCLS;

    const float NEG = -__builtin_inff();
    // classes handled by this lane: lane, lane+32, lane+64
    float x0 = src[lane];                                       // lane < 91 always
    float x1 = src[lane + 32];                                  // 32..63 < 91 always
    float x2 = (lane + 64) < NCLS ? src[lane + 64] : NEG;

    float m = fmaxf(x0, fmaxf(x1, x2));
    #pragma unroll
    for (int off = 16; off; off >>= 1) m = fmaxf(m, __shfl_xor(m, off, 32));

    float e0 = __expf(x0 - m);
    float e1 = __expf(x1 - m);
    float e2 = (lane + 64) < NCLS ? __expf(x2 - m) : 0.0f;

    float s = e0 + e1 + e2;
    #pragma unroll
    for (int off = 16; off; off >>= 1) s += __shfl_xor(s, off, 32);

    const float inv = 1.0f / s;
    float* dst = probs + (size_t)row * PK;
    dst[lane]      = e0 * inv;
    dst[lane + 32] = e1 * inv;
    dst[lane + 64] = (lane + 64) < NCLS ? e2 * inv : 0.0f;   // zero the K-pad
}

// ---------------------------------------------------------------------------
// Kernel 2: per-(b, q-tile, t-tile) 16x16 cost tile.
//   class cost via V_WMMA_F32_16X16X4_F32 against an in-register one-hot B
//   (exact gather: products are p*1 / p*0, zero-adds are lossless),
//   bbox L1 cost via VALU on the D-layout lanes.
// One wave32 per block; EXEC is always full (no divergence) as WMMA requires.
// ---------------------------------------------------------------------------
__global__ void hm_cost_tiles(const float* __restrict__ probs,
                              const float* __restrict__ pred_boxes,
                              const int*   __restrict__ tgt_labels,
                              const float* __restrict__ tgt_boxes,
                              float* __restrict__ out) {
    const int lane = threadIdx.x;          // 0..31
    const int t0   = blockIdx.x * 16;
    const int q0   = blockIdx.y * 16;
    const int b    = blockIdx.z;

    const int half = lane >> 4;            // 0: K pair {0,1} / rows 0-7 ; 1: {2,3} / rows 8-15
    const int lid  = lane & 15;

    // One-hot column data: target class id for column N = lid
    const int tcol  = t0 + lid;
    const int tidn  = tgt_labels[b * T + tcol];

    // A-matrix source row: probs[b*Q + q0 + lid][k0 + 2*half + {0,1}]
    const float* arow = probs + ((size_t)(b * Q + q0 + lid)) * PK + 2 * half;

    v8f acc = {};
    #pragma unroll
    for (int s = 0; s < PK / 4; ++s) {
        const int k0 = 4 * s;
        // 8-byte aligned b64 load of the two A elements this lane owns
        v2f A = *(const v2f*)(arow + k0);
        // B 4x16: lanes 0-15 carry K = k0+0,k0+1 ; lanes 16-31 carry K = k0+2,k0+3
        const int kk = k0 + 2 * half;
        v2f B;
        B.x = (tidn == kk)     ? 1.0f : 0.0f;
        B.y = (tidn == kk + 1) ? 1.0f : 0.0f;
        // (neg_a, A, neg_b, B, c_mod, C, reuse_a, reuse_b)
        acc = __builtin_amdgcn_wmma_f32_16x16x4_f32(false, A, false, B,
                                                    (short)0, acc, false, false);
    }

    // Target box for this lane's column
    const float* tb = tgt_boxes + ((size_t)(b * T + tcol)) * 4;
    const float tb0 = tb[0], tb1 = tb[1], tb2 = tb[2], tb3 = tb[3];

    // D layout: acc[r] is (M = r + 8*half, N = lid)
    const float* pb  = pred_boxes + ((size_t)(b * Q + q0 + 8 * half)) * 4;
    float*       orw = out + ((size_t)(b * Q + q0 + 8 * half)) * T + t0 + lid;

    #pragma unroll
    for (int r = 0; r < 8; ++r) {
        const float* p = pb + r * 4;
        const float l1 = fabsf(p[0] - tb0) + fabsf(p[1] - tb1) +
                         fabsf(p[2] - tb2) + fabsf(p[3] - tb3);
        // COST_BBOX * l1 + COST_CLASS * (-prob)
        orw[(size_t)r * T] = 5.0f * l1 - acc[r];
    }
}

// ---------------------------------------------------------------------------
extern "C" void kernel_launch(void* const* d_in, const int* in_sizes, int n_in,
                              void* d_out, int out_size, void* d_ws, size_t ws_size,
                              hipStream_t stream) {
    const float* pred_logits = (const float*)d_in[0];   // [BS, Q, NCLS] f32
    const float* pred_boxes  = (const float*)d_in[1];   // [BS, Q, 4]    f32
    const int*   tgt_labels  = (const int*)  d_in[2];   // [BS, T]       int
    const float* tgt_boxes   = (const float*)d_in[3];   // [BS, T, 4]    f32
    float*       out         = (float*)d_out;           // [BS, Q, T]    f32

    float* probs = (float*)d_ws;                        // [BS*Q, PK] = 6 MB

    // 1) softmax: 16384 rows, 8 rows (waves) per 256-thread block
    hm_softmax_rows<<<dim3((BS * Q) / 8), dim3(256), 0, stream>>>(pred_logits, probs);

    // 2) cost tiles: one wave32 per 16x16 (q,t) tile
    hm_cost_tiles<<<dim3(T / 16, Q / 16, BS), dim3(32), 0, stream>>>(
        probs, pred_boxes, tgt_labels, tgt_boxes, out);
}